// Flowformer_attention_3427383902645
// MI455X (gfx1250) — compile-verified
//

#include <hip/hip_runtime.h>
#include <hip/hip_bf16.h>

// ---------------------------------------------------------------------------
// Flowformer attention for MI455X (gfx1250), fp32 throughout.
// B=2, L=2048, D_MODEL=512, H=8, D=64.
// Matrix work: V_WMMA_F32_16X16X4_F32 (wave32). Chunk staging in the linear
// attention kernel uses GLOBAL_LOAD_ASYNC_TO_LDS_B128 (+ s_wait_asynccnt).
// ---------------------------------------------------------------------------

#define FF_B 2
#define FF_L 2048
#define FF_DM 512
#define FF_H 8
#define FF_D 64
#define FF_M (FF_B * FF_L)          // 4096 token rows
#define FF_EPS 1e-6f

typedef float v2f __attribute__((ext_vector_type(2)));
typedef float v8f __attribute__((ext_vector_type(8)));

__device__ __forceinline__ v8f wmma_f32(v2f a, v2f b, v8f c) {
  // D = A(16x4, f32) * B(4x16, f32) + C(16x16, f32)
  return __builtin_amdgcn_wmma_f32_16x16x4_f32(
      /*neg_a=*/false, a, /*neg_b=*/false, b,
      /*c_mod=*/(short)0, c, /*reuse_a=*/false, /*reuse_b=*/false);
}

__device__ __forceinline__ float fastrcp(float x) {
  return __builtin_amdgcn_rcpf(x);        // v_rcp_f32, ~1 ulp
}

__device__ __forceinline__ float sigmoidf_dev(float x) {
  return fastrcp(1.0f + __expf(-x));
}

// Async memory->LDS copy, 16B per lane, tracked by ASYNCcnt (gfx1250).
// ldsAddr: per-lane LDS byte address; gOff: per-lane unsigned byte offset
// added to the 64-bit scalar base.
__device__ __forceinline__ void async_load_b128(unsigned ldsAddr, unsigned gOff,
                                                const void* base) {
  asm volatile("global_load_async_to_lds_b128 %0, %1, %2 offset:0"
               :: "v"(ldsAddr), "v"(gOff), "s"(base)
               : "memory");
}
__device__ __forceinline__ void wait_async0() {
  asm volatile("s_wait_asynccnt 0x0" ::: "memory");
}

// ---------------------------------------------------------------------------
// Kernel 1: out = act(A @ W + bias).  A: [FF_M, 512] row-major, W: [512,512].
// permute==1 -> out stored as [B, H, L, D]; permute==0 -> row-major [FF_M,512].
// One wave computes a 32x32 block of C as 2x2 WMMA tiles (A/B fragments each
// reused twice: 8 loads per 4 WMMA). 8 waves per block.
// ---------------------------------------------------------------------------
__global__ __launch_bounds__(256)
void proj_gemm(const float* __restrict__ A, const float* __restrict__ W,
               const float* __restrict__ bias, float* __restrict__ out,
               int act, int permute) {
  const int lane = threadIdx.x & 31;
  const int wave = threadIdx.x >> 5;
  const int unit = blockIdx.x * 8 + wave;   // (4096/32)*(512/32) = 2048 units
  const int tr = unit >> 4;                 // 16 column units
  const int tc = unit & 15;

  const int rlane = lane & 15;              // row (A) / col (B,C) within tile
  const int koff  = (lane >> 4) * 2;        // K sub-offset per half-wave
  const int mbase = (lane >> 4) * 8;        // C/D: lanes 16-31 hold M=8..15

  const int row0 = tr * 32 + rlane;
  const int row1 = row0 + 16;
  const int col0 = tc * 32 + rlane;
  const int col1 = col0 + 16;

  v8f c00 = {}, c01 = {}, c10 = {}, c11 = {};
  const float* ap0 = A + (size_t)row0 * FF_DM + koff;
  const float* ap1 = A + (size_t)row1 * FF_DM + koff;
  const float* bp0 = W + (size_t)koff * FF_DM + col0;
  const float* bp1 = W + (size_t)koff * FF_DM + col1;
  for (int k0 = 0; k0 < FF_DM; k0 += 4) {
    v2f a0, a1, b0, b1;
    a0.x = ap0[k0];              a0.y = ap0[k0 + 1];
    a1.x = ap1[k0];              a1.y = ap1[k0 + 1];
    b0.x = bp0[k0 * FF_DM];      b0.y = bp0[(k0 + 1) * FF_DM];
    b1.x = bp1[k0 * FF_DM];      b1.y = bp1[(k0 + 1) * FF_DM];
    c00 = wmma_f32(a0, b0, c00);
    c01 = wmma_f32(a0, b1, c01);
    c10 = wmma_f32(a1, b0, c10);
    c11 = wmma_f32(a1, b1, c11);
  }

  const float bc0 = bias[col0];
  const float bc1 = bias[col1];
  v8f cs[4] = {c00, c01, c10, c11};
  for (int tile = 0; tile < 4; ++tile) {
    const int rowb = tr * 32 + ((tile >> 1) ? 16 : 0) + mbase;
    const int col  = (tile & 1) ? col1 : col0;
    const float bc = (tile & 1) ? bc1 : bc0;
    for (int r = 0; r < 8; ++r) {
      const int row = rowb + r;
      float v = cs[tile][r] + bc;
      if (act) v = sigmoidf_dev(v);
      if (permute) {
        const int b_ = row >> 11;        // / FF_L
        const int l_ = row & (FF_L - 1);
        const int h_ = col >> 6;         // / FF_D
        const int d_ = col & (FF_D - 1);
        out[(((size_t)(b_ * FF_H + h_) * FF_L) + l_) * FF_D + d_] = v;
      } else {
        out[(size_t)row * FF_DM + col] = v;
      }
    }
  }
}

// ---------------------------------------------------------------------------
// Kernel 2: flow-conservation statistics. One block (256 thr) per (b,h).
// Mutates q -> qq and v -> vv in place; writes alloc (sink_allocation) per l.
// ---------------------------------------------------------------------------
__global__ __launch_bounds__(256)
void flow_stats(float* __restrict__ qbase, const float* __restrict__ kbase,
                float* __restrict__ vbase,
                float* __restrict__ cumkbase, float* __restrict__ cumqbase,
                float* __restrict__ sibase, float* __restrict__ sobase,
                float* __restrict__ allocbase, float* __restrict__ scompbase) {
  const int bh = blockIdx.x;
  float*       q  = qbase   + (size_t)bh * FF_L * FF_D;
  const float* k  = kbase   + (size_t)bh * FF_L * FF_D;
  float*       v  = vbase   + (size_t)bh * FF_L * FF_D;
  float*       ck = cumkbase + (size_t)bh * FF_L * FF_D;
  float*       cq = cumqbase + (size_t)bh * FF_L * FF_D;
  float* siL    = sibase    + (size_t)bh * FF_L;
  float* soL    = sobase    + (size_t)bh * FF_L;
  float* allocL = allocbase + (size_t)bh * FF_L;
  float* scompL = scompbase + (size_t)bh * FF_L;

  const int t = threadIdx.x;
  const int lane = t & 31;
  const int wave = t >> 5;

  __shared__ float part[256];

  // Phase A: cumk = cumsum_l(k), cumq = cumsum_l(q)  (thread per d)
  if (t < 128) {
    const int d = t & 63;
    const float* src = (t < 64) ? k : q;
    float* dst = (t < 64) ? ck : cq;
    float run = 0.0f;
    for (int l = 0; l < FF_L; ++l) {
      run += src[l * FF_D + d];
      dst[l * FF_D + d] = run;
    }
  }
  __syncthreads();

  // Phase B: sink_incoming / source_outgoing (wave per token, shuffle reduce)
  for (int l = wave; l < FF_L; l += 8) {
    float s1 = 0.0f, s2 = 0.0f;
    for (int d = lane; d < FF_D; d += 32) {
      s1 += (q[l * FF_D + d] + FF_EPS) * (ck[l * FF_D + d] + FF_EPS);
      s2 += (k[l * FF_D + d] + FF_EPS) * (cq[l * FF_D + d] + FF_EPS);
    }
    for (int off = 16; off; off >>= 1) {
      s1 += __shfl_xor(s1, off);
      s2 += __shfl_xor(s2, off);
    }
    if (lane == 0) {
      const float n = (float)(l + 1);
      siL[l] = n * fastrcp(s1);   // sink_incoming (incl *normal)
      soL[l] = n * fastrcp(s2);   // source_outgoing (incl *normal)
    }
  }
  __syncthreads();

  // Phase C: cumsum of k*so and q*si (overwrites ck, cq)
  if (t < 128) {
    const int d = t & 63;
    float run = 0.0f;
    if (t < 64) {
      for (int l = 0; l < FF_L; ++l) {
        run += k[l * FF_D + d] * soL[l];
        ck[l * FF_D + d] = run;
      }
    } else {
      for (int l = 0; l < FF_L; ++l) {
        run += q[l * FF_D + d] * siL[l];
        cq[l * FF_D + d] = run;
      }
    }
  }
  __syncthreads();

  // Phase D: conserved sink/source -> alloc, src=exp(clip(conserved_source))
  for (int l = wave; l < FF_L; l += 8) {
    float s1 = 0.0f, s2 = 0.0f;
    for (int d = lane; d < FF_D; d += 32) {
      s1 += (q[l * FF_D + d] + FF_EPS) * (ck[l * FF_D + d] + FF_EPS);
      s2 += (k[l * FF_D + d] + FF_EPS) * (cq[l * FF_D + d] + FF_EPS);
    }
    for (int off = 16; off; off >>= 1) {
      s1 += __shfl_xor(s1, off);
      s2 += __shfl_xor(s2, off);
    }
    if (lane == 0) {
      const float rn = fastrcp((float)(l + 1));
      allocL[l] = sigmoidf_dev(s1 * rn);                // sink_allocation
      float cs = s2 * rn;
      cs = fminf(10.0f, fmaxf(-10.0f, cs));
      scompL[l] = __expf(cs);                           // src (temp)
    }
  }
  __syncthreads();

  // Phase E: inclusive scan of src over L; scomp = src/cumsum(src)*normal
  float loc[8];
  float run = 0.0f;
  for (int i = 0; i < 8; ++i) {
    loc[i] = scompL[t * 8 + i];
    run += loc[i];
  }
  part[t] = run;
  __syncthreads();
  for (int off = 1; off < 256; off <<= 1) {
    const float val = part[t];
    const float add = (t >= off) ? part[t - off] : 0.0f;
    __syncthreads();
    part[t] = val + add;
    __syncthreads();
  }
  float cacc = (t == 0) ? 0.0f : part[t - 1];
  for (int i = 0; i < 8; ++i) {
    cacc += loc[i];
    const int l = t * 8 + i;
    scompL[l] = loc[i] * (float)(l + 1) * fastrcp(cacc); // source_competition
  }
  __syncthreads();

  // Phase F: qq = q*si/normal, vv = v*scomp  (in place)
  for (int idx = t; idx < FF_L * FF_D; idx += 256) {
    const int l = idx >> 6;
    q[idx] = q[idx] * siL[l] * fastrcp((float)(l + 1));
    v[idx] = v[idx] * scompL[l];
  }
}

// ---------------------------------------------------------------------------
// Kernel 3: chunked causal linear attention, chunk C=64, kv state in LDS.
// One block (8 waves) per (b,h); 32 sequential chunks.
//   S  = qq_c @ K_c^T (causal-masked within chunk)
//   X  = (S @ VV_c + qq_c @ kv_prev) * alloc
//   kv += K_c^T @ VV_c
// K/VV chunks staged with GLOBAL_LOAD_ASYNC_TO_LDS_B128 (ASYNCcnt tracked).
// X written as [B, L, H, D] (row-major [4096, 512]) for the output GEMM.
// ---------------------------------------------------------------------------
__global__ __launch_bounds__(256)
void linear_attn(const float* __restrict__ qqbase, const float* __restrict__ kbase,
                 const float* __restrict__ vvbase, const float* __restrict__ allocbase,
                 float* __restrict__ X) {
  const int bh = blockIdx.x;
  const int b_ = bh >> 3;
  const int h_ = bh & 7;
  const float* qq = qqbase   + (size_t)bh * FF_L * FF_D;
  const float* kk = kbase    + (size_t)bh * FF_L * FF_D;
  const float* vv = vvbase   + (size_t)bh * FF_L * FF_D;
  const float* al = allocbase + (size_t)bh * FF_L;

  __shared__ float kv[64 * 64];   // running state (16 KB)
  __shared__ float sS[64 * 64];   // masked scores
  __shared__ float sK[64 * 64];   // k chunk
  __shared__ float sV[64 * 64];   // vv chunk            total 64 KB LDS

  const int t = threadIdx.x;
  const int lane = t & 31;
  const int wave = t >> 5;
  const int rlane = lane & 15;
  const int koff  = (lane >> 4) * 2;
  const int mbase = (lane >> 4) * 8;

  const unsigned sKlds = (unsigned)(uintptr_t)&sK[0];
  const unsigned sVlds = (unsigned)(uintptr_t)&sV[0];

  for (int i = t; i < 4096; i += 256) kv[i] = 0.0f;
  __syncthreads();

  for (int ch = 0; ch < 32; ++ch) {
    const int l0 = ch * 64;
    // stage chunk of K and VV into LDS via async b128 (16B per lane per issue)
    {
      const unsigned gbase = (unsigned)(l0 * FF_D * 4);
      for (unsigned i = (unsigned)t * 16u; i < 64u * 64u * 4u; i += 256u * 16u) {
        async_load_b128(sKlds + i, gbase + i, kk);
        async_load_b128(sVlds + i, gbase + i, vv);
      }
      wait_async0();
    }
    __syncthreads();

    // ---- step 2: S = qq @ K^T, causal mask on spill ----
    for (int tt = wave; tt < 16; tt += 8) {
      const int tr = tt >> 2, tc = tt & 3;
      v8f c = {};
      const float* ap = qq + (size_t)(l0 + tr * 16 + rlane) * FF_D + koff;
      const float* bp = &sK[(tc * 16 + rlane) * 64 + koff];   // K^T[kk][j] = K[j][kk]
      for (int k0 = 0; k0 < 64; k0 += 4) {
        v2f a, b;
        a.x = ap[k0];       a.y = ap[k0 + 1];
        b.x = bp[k0];       b.y = bp[k0 + 1];
        c = wmma_f32(a, b, c);
      }
      for (int r = 0; r < 8; ++r) {
        const int m = tr * 16 + mbase + r;     // local row l
        const int n = tc * 16 + rlane;         // local col j
        sS[m * 64 + n] = (n <= m) ? c[r] : 0.0f;
      }
    }
    __syncthreads();

    // ---- step 3: X = S @ VV + qq @ kv_prev, scale by alloc, store ----
    for (int tt = wave; tt < 16; tt += 8) {
      const int tr = tt >> 2, tc = tt & 3;
      v8f c = {};
      {
        const float* ap = &sS[(tr * 16 + rlane) * 64 + koff];
        const float* bp = &sV[(koff) * 64 + tc * 16 + rlane];
        for (int k0 = 0; k0 < 64; k0 += 4) {
          v2f a, b;
          a.x = ap[k0];           a.y = ap[k0 + 1];
          b.x = bp[k0 * 64];      b.y = bp[(k0 + 1) * 64];
          c = wmma_f32(a, b, c);
        }
      }
      {
        const float* ap = qq + (size_t)(l0 + tr * 16 + rlane) * FF_D + koff;
        const float* bp = &kv[(koff) * 64 + tc * 16 + rlane];
        for (int k0 = 0; k0 < 64; k0 += 4) {
          v2f a, b;
          a.x = ap[k0];           a.y = ap[k0 + 1];
          b.x = bp[k0 * 64];      b.y = bp[(k0 + 1) * 64];
          c = wmma_f32(a, b, c);
        }
      }
      for (int r = 0; r < 8; ++r) {
        const int m = tr * 16 + mbase + r;
        const int lg = l0 + m;
        const int n = tc * 16 + rlane;
        const float val = c[r] * al[lg];
        X[((size_t)(b_ * FF_L + lg) * FF_H + h_) * FF_D + n] = val;
      }
    }
    __syncthreads();

    // ---- step 4: kv += K^T @ VV ----
    for (int tt = wave; tt < 16; tt += 8) {
      const int tr = tt >> 2, tc = tt & 3;
      v8f c = {};
      const float* apk = &sK[(koff) * 64 + tr * 16 + rlane];  // K^T[d][j] = K[j][d]
      const float* bp  = &sV[(koff) * 64 + tc * 16 + rlane];
      for (int k0 = 0; k0 < 64; k0 += 4) {
        v2f a, b;
        a.x = apk[k0 * 64];     a.y = apk[(k0 + 1) * 64];
        b.x = bp[k0 * 64];      b.y = bp[(k0 + 1) * 64];
        c = wmma_f32(a, b, c);
      }
      for (int r = 0; r < 8; ++r) {
        const int m = tr * 16 + mbase + r;
        const int n = tc * 16 + rlane;
        kv[m * 64 + n] += c[r];
      }
    }
    __syncthreads();
  }
}

// ---------------------------------------------------------------------------
// Launch
// ---------------------------------------------------------------------------
extern "C" void kernel_launch(void* const* d_in, const int* in_sizes, int n_in,
                              void* d_out, int out_size, void* d_ws, size_t ws_size,
                              hipStream_t stream) {
  (void)in_sizes; (void)n_in; (void)out_size; (void)ws_size;

  const float* queries = (const float*)d_in[0];
  const float* keys    = (const float*)d_in[1];
  const float* values  = (const float*)d_in[2];
  const float* Wq = (const float*)d_in[3];
  const float* bq = (const float*)d_in[4];
  const float* Wk = (const float*)d_in[5];
  const float* bk = (const float*)d_in[6];
  const float* Wv = (const float*)d_in[7];
  const float* bv = (const float*)d_in[8];
  const float* Wo = (const float*)d_in[9];
  const float* bo = (const float*)d_in[10];

  float* ws = (float*)d_ws;
  const size_t SZ = (size_t)FF_B * FF_H * FF_L * FF_D;  // 2,097,152 floats
  float* qb   = ws;
  float* kb   = ws + 1 * SZ;
  float* vb   = ws + 2 * SZ;
  float* cumk = ws + 3 * SZ;
  float* cumq = ws + 4 * SZ;
  float* Xb   = ws + 5 * SZ;
  float* si    = ws + 6 * SZ;
  float* so    = si + (size_t)FF_B * FF_H * FF_L;
  float* alloc = so + (size_t)FF_B * FF_H * FF_L;
  float* scomp = alloc + (size_t)FF_B * FF_H * FF_L;

  const dim3 blk(256);
  const dim3 gemmGrid((FF_M / 32) * (FF_DM / 32) / 8);  // 256 blocks
  const dim3 bhGrid(FF_B * FF_H);                       // 16 blocks

  // Q/K projections with fused sigmoid, V projection plain, to [B,H,L,D]
  proj_gemm<<<gemmGrid, blk, 0, stream>>>(queries, Wq, bq, qb, 1, 1);
  proj_gemm<<<gemmGrid, blk, 0, stream>>>(keys,    Wk, bk, kb, 1, 1);
  proj_gemm<<<gemmGrid, blk, 0, stream>>>(values,  Wv, bv, vb, 0, 1);

  // Flow conservation stats; mutates qb -> qq, vb -> vv, produces alloc
  flow_stats<<<bhGrid, blk, 0, stream>>>(qb, kb, vb, cumk, cumq, si, so, alloc, scomp);

  // Chunked causal linear attention -> Xb as [B, L, H*D]
  linear_attn<<<bhGrid, blk, 0, stream>>>(qb, kb, vb, alloc, Xb);

  // Output projection: d_out = Xb @ Wo + bo
  proj_gemm<<<gemmGrid, blk, 0, stream>>>(Xb, Wo, bo, (float*)d_out, 0, 0);
}